// Decoder_27951647162549
// MI455X (gfx1250) — compile-verified
//
#include <hip/hip_runtime.h>
#include <hip/hip_bf16.h>
#include <math.h>

typedef __attribute__((ext_vector_type(2))) float v2f;
typedef __attribute__((ext_vector_type(8))) float v8f;

#define HDIM  512
#define EDIM  256
#define BATCH 64
#define TLEN  512
#define VOCAB 32000

__device__ __forceinline__ v8f zero8() {
  v8f z;
  #pragma unroll
  for (int i = 0; i < 8; ++i) z[i] = 0.0f;
  return z;
}

// Async global->LDS copy, 16 bytes per lane. CDNA5 GLOBAL_LOAD_ASYNC_TO_LDS_B128:
// VDST = VGPR holding LDS byte address (flat-shared addr low 32 bits == LDS offset),
// VADDR = 64-bit global address. Tracked by ASYNCcnt.
__device__ __forceinline__ void async_b128(const float* lds_dst, const float* gsrc) {
  asm volatile("global_load_async_to_lds_b128 %0, %1, off"
               :: "v"((unsigned int)(unsigned long long)(uintptr_t)lds_dst),
                  "v"((unsigned long long)(uintptr_t)gsrc)
               : "memory");
}
__device__ __forceinline__ void wait_async0() {
  asm volatile("s_wait_asynccnt 0x0" ::: "memory");
}

// Load a 16x4 fp32 WMMA fragment (A or B) from LDS.
// ISA layout (32-bit A 16x4): lanes 0-15 hold {K=k0,k0+1}, lanes 16-31 hold {K=k0+2,k0+3},
// row = row0 + (lane & 15). W is staged as sW[n][k] so B fragments use identical indexing.
__device__ __forceinline__ v2f ld_frag(const float* s, int row0, int stride, int k0, int lane) {
  int r = row0 + (lane & 15);
  int k = k0 + ((lane >> 4) << 1);
  v2f f;
  f.x = s[r * stride + k];
  f.y = s[r * stride + k + 1];
  return f;
}

// ---------------------------------------------------------------------------
// Generic fp32 WMMA GEMM:  C[64 x N] = A[64 x K] @ W[N x K]^T + bias[N]
// grid.x = N/64, block = 128 threads (4 waves). Each wave computes a 16x64 strip.
// Double-buffered LDS, async global->LDS staging overlapped with WMMA compute.
// ---------------------------------------------------------------------------
__global__ __launch_bounds__(128)
void gemm_bias_kernel(const float* __restrict__ A, int lda,
                      const float* __restrict__ W, int ldw,
                      const float* __restrict__ bias,
                      float* __restrict__ C, int ldc,
                      int K) {
  __shared__ float sA[2][64 * 68];
  __shared__ float sW[2][64 * 68];
  const int tid  = threadIdx.x;
  const int lane = tid & 31;
  const int wave = tid >> 5;
  const int n0   = blockIdx.x * 64;
  const int m0   = wave * 16;

  auto stage = [&](int buf, int kk) {
    #pragma unroll
    for (int idx = tid; idx < 64 * 16; idx += 128) {
      int r = idx >> 4, s = idx & 15;
      async_b128(sA[buf] + r * 68 + s * 4, A + (size_t)r * lda + kk + s * 4);
      async_b128(sW[buf] + r * 68 + s * 4, W + (size_t)(n0 + r) * ldw + kk + s * 4);
    }
  };

  v8f acc[4];
  #pragma unroll
  for (int ct = 0; ct < 4; ++ct) acc[ct] = zero8();

  stage(0, 0);
  wait_async0();
  __syncthreads();

  int cur = 0;
  for (int kk = 0; kk < K; kk += 64) {
    if (kk + 64 < K) stage(cur ^ 1, kk + 64);   // prefetch next chunk (other buffer)

    const float* cA = sA[cur];
    const float* cW = sW[cur];
    #pragma unroll
    for (int k4 = 0; k4 < 16; ++k4) {
      v2f a = ld_frag(cA, m0, 68, k4 * 4, lane);
      #pragma unroll
      for (int ct = 0; ct < 4; ++ct) {
        v2f b = ld_frag(cW, ct * 16, 68, k4 * 4, lane);
        acc[ct] = __builtin_amdgcn_wmma_f32_16x16x4_f32(
            false, a, false, b, (short)0, acc[ct], false, false);
      }
    }
    wait_async0();     // my async writes into the other buffer are complete
    __syncthreads();   // everyone done reading cur / writing next
    cur ^= 1;
  }

  // C/D layout: VGPR vr, lanes 0-15 -> row m0+vr, lanes 16-31 -> row m0+8+vr, col = (lane&15)
  const int rbase = m0 + ((lane >> 4) << 3);
  #pragma unroll
  for (int ct = 0; ct < 4; ++ct) {
    int n = n0 + ct * 16 + (lane & 15);
    float bz = bias[n];
    #pragma unroll
    for (int vr = 0; vr < 8; ++vr)
      C[(size_t)(rbase + vr) * ldc + n] = acc[ct][vr] + bz;
  }
}

// ---------------------------------------------------------------------------
// Fused attention energy + score:
// scores[b,t] = sum_n v[n] * tanh( enc[b,t,:]·W_enc[n,:] + dec[b,n] )
// grid.x = B*T/64 = 512; block handles 64 consecutive t of one b.
// Full 64x512 enc tile staged once (async, direct-to-LDS) in 132KB of LDS.
// ---------------------------------------------------------------------------
__global__ __launch_bounds__(128)
void attn_energy_kernel(const float* __restrict__ enc,   // [B,T,H]
                        const float* __restrict__ Wattn, // [H,2H], enc part = cols 0..H
                        const float* __restrict__ dec,   // [B,H]
                        const float* __restrict__ v,     // [H]
                        float* __restrict__ scores) {    // [B,T]
  __shared__ float sA[64 * 516];  // padded stride: conflict-free, 16B-aligned rows
  __shared__ float sW[64 * 68];
  const int tid  = threadIdx.x;
  const int lane = tid & 31;
  const int wave = tid >> 5;
  const int mb   = blockIdx.x;
  const int b    = mb >> 3;
  const int t0   = (mb & 7) * 64;
  const int m0   = wave * 16;
  const int half8 = (lane >> 4) << 3;

  const float* Ab = enc + ((size_t)b * TLEN + t0) * HDIM;
  for (int idx = tid; idx < 64 * 128; idx += 128) {
    int r = idx >> 7, s = idx & 127;
    async_b128(sA + r * 516 + s * 4, Ab + (size_t)r * HDIM + s * 4);
  }
  wait_async0();
  __syncthreads();

  v8f rowacc = zero8();

  for (int nt = 0; nt < 8; ++nt) {
    const int n0 = nt * 64;
    v8f acc[4];
    #pragma unroll
    for (int ct = 0; ct < 4; ++ct) acc[ct] = zero8();

    for (int kk = 0; kk < HDIM; kk += 64) {
      for (int idx = tid; idx < 64 * 16; idx += 128) {
        int r = idx >> 4, s = idx & 15;
        float4 d = *(const float4*)(Wattn + (size_t)(n0 + r) * (2 * HDIM) + kk + s * 4);
        *(float4*)(sW + r * 68 + s * 4) = d;
      }
      __syncthreads();
      #pragma unroll
      for (int k4 = 0; k4 < 16; ++k4) {
        v2f a = ld_frag(sA, m0, 516, kk + k4 * 4, lane);
        #pragma unroll
        for (int ct = 0; ct < 4; ++ct) {
          v2f bf = ld_frag(sW, ct * 16, 68, k4 * 4, lane);
          acc[ct] = __builtin_amdgcn_wmma_f32_16x16x4_f32(
              false, a, false, bf, (short)0, acc[ct], false, false);
        }
      }
      __syncthreads();
    }

    // tanh(+dec) * v, accumulate per-row partial sums (rows fixed per lane/vr)
    #pragma unroll
    for (int ct = 0; ct < 4; ++ct) {
      int n = n0 + ct * 16 + (lane & 15);
      float dv = dec[b * HDIM + n];
      float vv = v[n];
      #pragma unroll
      for (int vr = 0; vr < 8; ++vr)
        rowacc[vr] += tanhf(acc[ct][vr] + dv) * vv;
    }
  }

  // reduce the 16 column-lanes within each half-wave
  #pragma unroll
  for (int m = 8; m >= 1; m >>= 1) {
    #pragma unroll
    for (int vr = 0; vr < 8; ++vr)
      rowacc[vr] += __shfl_xor(rowacc[vr], m, 32);
  }
  if ((lane & 15) == 0) {
    #pragma unroll
    for (int vr = 0; vr < 8; ++vr)
      scores[(size_t)b * TLEN + t0 + m0 + half8 + vr] = rowacc[vr];
  }
}

// ---------------------------------------------------------------------------
__global__ __launch_bounds__(256)
void embed_kernel(const int* __restrict__ tgt, const float* __restrict__ table,
                  float* __restrict__ emb) {
  int idx = blockIdx.x * 256 + threadIdx.x;       // B*E = 16384
  int b = idx >> 8, e = idx & 255;
  emb[idx] = table[(size_t)tgt[b] * EDIM + e];
}

__global__ __launch_bounds__(256)
void gru_gate_kernel(const float* __restrict__ gi, const float* __restrict__ gh,
                     const float* __restrict__ hprev,
                     float* __restrict__ out_hidden,   // [B,H] in d_out
                     float* __restrict__ out2, int out2_stride) {
  int idx = blockIdx.x * 256 + threadIdx.x;       // B*H = 32768
  int b = idx >> 9, j = idx & 511;
  const float* gib = gi + (size_t)b * 1536;
  const float* ghb = gh + (size_t)b * 1536;
  float r = 1.0f / (1.0f + expf(-(gib[j] + ghb[j])));
  float z = 1.0f / (1.0f + expf(-(gib[512 + j] + ghb[512 + j])));
  float n = tanhf(gib[1024 + j] + r * ghb[1024 + j]);
  float h = (1.0f - z) * n + z * hprev[(size_t)b * HDIM + j];
  out_hidden[(size_t)b * HDIM + j] = h;
  out2[(size_t)b * out2_stride + j] = h;
}

__global__ __launch_bounds__(256)
void softmax_kernel(const float* __restrict__ scores, float* __restrict__ attn) {
  __shared__ float red[256];
  int b = blockIdx.x, tid = threadIdx.x;
  float x0 = scores[(size_t)b * TLEN + tid];
  float x1 = scores[(size_t)b * TLEN + tid + 256];
  red[tid] = fmaxf(x0, x1);
  __syncthreads();
  for (int s = 128; s > 0; s >>= 1) {
    if (tid < s) red[tid] = fmaxf(red[tid], red[tid + s]);
    __syncthreads();
  }
  float m = red[0];
  __syncthreads();
  float e0 = expf(x0 - m), e1 = expf(x1 - m);
  red[tid] = e0 + e1;
  __syncthreads();
  for (int s = 128; s > 0; s >>= 1) {
    if (tid < s) red[tid] += red[tid + s];
    __syncthreads();
  }
  float inv = 1.0f / red[0];
  attn[(size_t)b * TLEN + tid]       = e0 * inv;
  attn[(size_t)b * TLEN + tid + 256] = e1 * inv;
}

__global__ __launch_bounds__(256)
void context_kernel(const float* __restrict__ enc, const float* __restrict__ attn,
                    float* __restrict__ combined) {   // writes cols [H,2H) of [B,2H]
  __shared__ float sw[TLEN];
  int b = blockIdx.x, tid = threadIdx.x;
  sw[tid]       = attn[(size_t)b * TLEN + tid];
  sw[tid + 256] = attn[(size_t)b * TLEN + tid + 256];
  __syncthreads();
  float a0 = 0.0f, a1 = 0.0f;
  const float* e = enc + (size_t)b * TLEN * HDIM;
  for (int t = 0; t < TLEN; ++t) {
    float w = sw[t];
    a0 += w * e[(size_t)t * HDIM + tid];
    a1 += w * e[(size_t)t * HDIM + tid + 256];
  }
  combined[(size_t)b * 1024 + HDIM + tid]       = a0;
  combined[(size_t)b * 1024 + HDIM + tid + 256] = a1;
}

// ---------------------------------------------------------------------------
extern "C" void kernel_launch(void* const* d_in, const int* in_sizes, int n_in,
                              void* d_out, int out_size, void* d_ws, size_t ws_size,
                              hipStream_t stream) {
  const float* enc    = (const float*)d_in[0];   // [B,T,H]
  const int*   tgt    = (const int*)  d_in[1];   // [B]
  const float* hidden = (const float*)d_in[2];   // [L,B,H]
  const float* table  = (const float*)d_in[3];   // [V,E]
  const float* W_ih0  = (const float*)d_in[4];
  const float* W_hh0  = (const float*)d_in[5];
  const float* b_ih0  = (const float*)d_in[6];
  const float* b_hh0  = (const float*)d_in[7];
  const float* W_ih1  = (const float*)d_in[8];
  const float* W_hh1  = (const float*)d_in[9];
  const float* b_ih1  = (const float*)d_in[10];
  const float* b_hh1  = (const float*)d_in[11];
  const float* W_attn = (const float*)d_in[12];  // [H,2H]
  const float* b_attn = (const float*)d_in[13];
  const float* v      = (const float*)d_in[14];
  const float* W_out  = (const float*)d_in[15];  // [V,2H]
  const float* b_out  = (const float*)d_in[16];

  float* out = (float*)d_out;
  float* logits  = out;                                    // [B,V]
  float* hid_out = out + (size_t)BATCH * VOCAB;            // [L,B,H]
  float* attn_w  = hid_out + (size_t)2 * BATCH * HDIM;     // [B,T]

  float* w = (float*)d_ws;
  float* emb      = w;                    // 64*256
  float* gi       = emb + 16384;          // 64*1536
  float* gh       = gi + 98304;           // 64*1536
  float* h0       = gh + 98304;           // 64*512
  float* dec      = h0 + 32768;           // 64*512
  float* scores   = dec + 32768;          // 64*512
  float* combined = scores + 32768;       // 64*1024  (h1 | context)

  // 1. embedding gather
  embed_kernel<<<64, 256, 0, stream>>>(tgt, table, emb);

  // 2. GRU layer 0
  gemm_bias_kernel<<<24, 128, 0, stream>>>(emb, EDIM, W_ih0, EDIM, b_ih0, gi, 1536, EDIM);
  gemm_bias_kernel<<<24, 128, 0, stream>>>(hidden, HDIM, W_hh0, HDIM, b_hh0, gh, 1536, HDIM);
  gru_gate_kernel<<<128, 256, 0, stream>>>(gi, gh, hidden, hid_out, h0, HDIM);

  // 3. GRU layer 1 (h1 also written into combined[:, 0:H], row stride 2H)
  gemm_bias_kernel<<<24, 128, 0, stream>>>(h0, HDIM, W_ih1, HDIM, b_ih1, gi, 1536, HDIM);
  gemm_bias_kernel<<<24, 128, 0, stream>>>(hidden + (size_t)BATCH * HDIM, HDIM,
                                           W_hh1, HDIM, b_hh1, gh, 1536, HDIM);
  gru_gate_kernel<<<128, 256, 0, stream>>>(gi, gh, hidden + (size_t)BATCH * HDIM,
                                           hid_out + (size_t)BATCH * HDIM, combined, 1024);

  // 4. decoder projection: dec = h1 @ W_dec^T + b_attn  (W_dec = W_attn[:, H:])
  gemm_bias_kernel<<<8, 128, 0, stream>>>(combined, 1024, W_attn + HDIM, 2 * HDIM,
                                          b_attn, dec, HDIM, HDIM);

  // 5. fused energy/tanh/v-dot -> scores ; softmax -> attn weights (output)
  attn_energy_kernel<<<512, 128, 0, stream>>>(enc, W_attn, dec, v, scores);
  softmax_kernel<<<64, 256, 0, stream>>>(scores, attn_w);

  // 6. context -> combined[:, H:2H]
  context_kernel<<<64, 256, 0, stream>>>(enc, attn_w, combined);

  // 7. logits = combined @ W_out^T + b_out
  gemm_bias_kernel<<<500, 128, 0, stream>>>(combined, 1024, W_out, 2 * HDIM,
                                            b_out, logits, VOCAB, 2 * HDIM);
}